// GCN_81681688035404
// MI455X (gfx1250) — compile-verified
//
#include <hip/hip_runtime.h>
#include <hip/hip_bf16.h>

#define N_NODES 100000
#define IN_C    256
#define HID_C   128
#define OUT_C   2
#define E_EDGES 3200000

typedef __attribute__((ext_vector_type(2))) float v2f;
typedef __attribute__((ext_vector_type(8))) float v8f;

// ---------------------------------------------------------------------------
// Phase 1: degree / norm precompute
// ---------------------------------------------------------------------------

__global__ void k_init_deg(float* __restrict__ deg) {
    int i = blockIdx.x * blockDim.x + threadIdx.x;
    if (i < N_NODES) deg[i] = 1.0f;   // self-loop weight
}

__global__ void k_accum_deg(const long long* __restrict__ ei,
                            const float* __restrict__ w,
                            float* __restrict__ deg) {
    int e = blockIdx.x * blockDim.x + threadIdx.x;
    if (e < E_EDGES) {
        int d = (int)ei[(long long)E_EDGES + e];   // dst row of edge_index
        atomicAdd(deg + d, w[e]);
    }
}

__global__ void k_dinv(float* __restrict__ deg) {
    int i = blockIdx.x * blockDim.x + threadIdx.x;
    if (i < N_NODES) {
        float dv = deg[i];
        deg[i] = (dv > 0.0f) ? rsqrtf(dv) : 0.0f;
    }
}

__global__ void k_edge_norm(const long long* __restrict__ ei,
                            const float* __restrict__ w,
                            const float* __restrict__ dinv,
                            float* __restrict__ nrm,
                            int* __restrict__ s32,
                            int* __restrict__ d32) {
    int e = blockIdx.x * blockDim.x + threadIdx.x;
    if (e < E_EDGES) {
        int s = (int)ei[e];
        int d = (int)ei[(long long)E_EDGES + e];
        s32[e] = s;
        d32[e] = d;
        nrm[e] = dinv[s] * w[e] * dinv[d];
    }
}

// ---------------------------------------------------------------------------
// Phase 2: h = x @ W1 via V_WMMA_F32_16X16X4_F32 (exact fp32)
// grid = (ceil(6250/16), 4), block = 256 (8 waves).
// Each wave computes a 2x2 group of 16x16 tiles (32 rows x 32 cols):
// every A fragment feeds 2 WMMAs and every B fragment feeds 2 WMMAs
// -> 0.5 global + 0.5 LDS ops per WMMA. Block covers 256 rows x 32 cols;
// the two 256x16 W1 column panels (32 KB) are staged in LDS.
// ---------------------------------------------------------------------------

__global__ __launch_bounds__(256) void k_gemm1_wmma(const float* __restrict__ x,
                                                    const float* __restrict__ W1,
                                                    float* __restrict__ h) {
    // LDS B panels: [panel][k][n], stride 16 floats.
    // B-fragment read: lanes 0-15 hit banks (k*16+L)%64, lanes 16-31 hit
    // ((k+2)*16+L)%64 = +32 mod 64 -> disjoint 16-bank groups, conflict-free.
    __shared__ float bs[2 * IN_C * 16];

    const int n0  = blockIdx.y * 32;
    const int tid = threadIdx.x;

    // Cooperative panel load: thread t loads W1 row t, cols n0..n0+31 (128B).
    {
        const float4* srcp = (const float4*)(W1 + tid * HID_C + n0);
        float4* d0 = (float4*)(bs + tid * 16);
        float4* d1 = (float4*)(bs + IN_C * 16 + tid * 16);
        d0[0] = srcp[0]; d0[1] = srcp[1]; d0[2] = srcp[2]; d0[3] = srcp[3];
        d1[0] = srcp[4]; d1[1] = srcp[5]; d1[2] = srcp[6]; d1[3] = srcp[7];
    }
    __syncthreads();

    const int wave   = tid >> 5;
    const int lane   = tid & 31;
    const int mtile0 = blockIdx.x * 16 + wave * 2;    // wave owns tiles {m0, m0+1}
    // 6250 total M-tiles is even, so each wave's pair is uniformly valid/invalid
    if (mtile0 * 16 >= N_NODES) return;               // wave-uniform branch

    const int lan16 = lane & 15;
    const int koff  = (lane >> 4) * 2;                // K split: lanes16-31 -> +2
    const float* arow0 = x + (mtile0 * 16 + lan16) * IN_C + koff;
    const float* arow1 = arow0 + 16 * IN_C;
    const float* bcol0 = bs + koff * 16 + lan16;
    const float* bcol1 = bcol0 + IN_C * 16;

    v8f acc00 = {}, acc01 = {}, acc10 = {}, acc11 = {};
    #pragma unroll 4
    for (int k0 = 0; k0 < IN_C; k0 += 4) {
        v2f a0, a1, b0, b1;                           // A 16x4: V0=K{0|2}, V1=K{1|3}
        a0.x = arow0[k0];  a0.y = arow0[k0 + 1];      // contiguous 8B -> b64 loads
        a1.x = arow1[k0];  a1.y = arow1[k0 + 1];
        b0.x = bcol0[k0 * 16];  b0.y = bcol0[k0 * 16 + 16];
        b1.x = bcol1[k0 * 16];  b1.y = bcol1[k0 * 16 + 16];
        acc00 = __builtin_amdgcn_wmma_f32_16x16x4_f32(
                    false, a0, false, b0, (short)0, acc00, false, false);
        acc01 = __builtin_amdgcn_wmma_f32_16x16x4_f32(
                    false, a0, false, b1, (short)0, acc01, false, false);
        acc10 = __builtin_amdgcn_wmma_f32_16x16x4_f32(
                    false, a1, false, b0, (short)0, acc10, false, false);
        acc11 = __builtin_amdgcn_wmma_f32_16x16x4_f32(
                    false, a1, false, b1, (short)0, acc11, false, false);
    }

    // D layout: VGPR r -> M = r + 8*(lane>=16), N = lane&15
    const int col = n0 + lan16;
    const int rb0 = mtile0 * 16 + (lane >> 4) * 8;
    const int rb1 = rb0 + 16;
    #pragma unroll
    for (int r = 0; r < 8; ++r) {
        h[(rb0 + r) * HID_C + col]      = acc00[r];
        h[(rb0 + r) * HID_C + col + 16] = acc01[r];
        h[(rb1 + r) * HID_C + col]      = acc10[r];
        h[(rb1 + r) * HID_C + col + 16] = acc11[r];
    }
}

// ---------------------------------------------------------------------------
// Phase 3: aggregation layer 1 (self-loop init + edge scatter), bias+ReLU
// ---------------------------------------------------------------------------

// agg[i] = dinv[i]^2 * h[i]   (self-loop message; also initializes agg)
__global__ void k_selfinit1(const float* __restrict__ h,
                            const float* __restrict__ dinv,
                            float* __restrict__ agg) {
    int t = blockIdx.x * blockDim.x + threadIdx.x;   // N*32 threads, float4 each
    if (t >= N_NODES * 32) return;
    int n = t >> 5;
    int c = (t & 31) * 4;
    float dv = dinv[n];
    float nm = dv * dv;
    float4 hv = *(const float4*)(h + n * HID_C + c);
    float4 o  = make_float4(nm * hv.x, nm * hv.y, nm * hv.z, nm * hv.w);
    *(float4*)(agg + n * HID_C + c) = o;
}

// One wave per edge: 32 lanes x float4 = 128-channel row; 4 atomics/lane.
// h and agg are both L2-resident (51.2 MB each, 192 MB L2); atomics are
// no-return global_atomic_add_f32 -> STOREcnt, fire-and-forget.
__global__ void k_scatter1(const int* __restrict__ s32,
                           const int* __restrict__ d32,
                           const float* __restrict__ nrm,
                           const float* __restrict__ h,
                           float* __restrict__ agg) {
    int t = blockIdx.x * blockDim.x + threadIdx.x;
    int e = t >> 5;
    if (e >= E_EDGES) return;
    int lane = t & 31;
    int s = s32[e];
    int d = d32[e];
    float nm = nrm[e];
    float4 hv = *(const float4*)(h + s * HID_C + lane * 4);
    float* ap = agg + d * HID_C + lane * 4;
    atomicAdd(ap + 0, nm * hv.x);
    atomicAdd(ap + 1, nm * hv.y);
    atomicAdd(ap + 2, nm * hv.z);
    atomicAdd(ap + 3, nm * hv.w);
}

// in-place: agg = relu(agg + b1)
__global__ void k_bias_relu(float* __restrict__ agg,
                            const float* __restrict__ b1) {
    int t = blockIdx.x * blockDim.x + threadIdx.x;
    if (t >= N_NODES * 32) return;
    int n = t >> 5;
    int c = (t & 31) * 4;
    float4 v = *(const float4*)(agg + n * HID_C + c);
    v.x = fmaxf(v.x + b1[c + 0], 0.0f);
    v.y = fmaxf(v.y + b1[c + 1], 0.0f);
    v.z = fmaxf(v.z + b1[c + 2], 0.0f);
    v.w = fmaxf(v.w + b1[c + 3], 0.0f);
    *(float4*)(agg + n * HID_C + c) = v;
}

// ---------------------------------------------------------------------------
// Phase 4: g2 = h2 @ W2 (K=128, N=2 -> VALU dot; W2 panel in LDS),
//          out = b2 + dinv^2*g2 + edge scatter
// ---------------------------------------------------------------------------

__global__ __launch_bounds__(256) void k_gemm2(const float* __restrict__ h2,
                                               const float* __restrict__ W2,
                                               float* __restrict__ g2) {
    __shared__ float w2s[HID_C * OUT_C];             // 256 floats
    w2s[threadIdx.x] = W2[threadIdx.x];
    __syncthreads();
    int i = blockIdx.x * blockDim.x + threadIdx.x;
    if (i >= N_NODES) return;
    const float4* hp = (const float4*)(h2 + i * HID_C);
    float a0 = 0.0f, a1 = 0.0f;
    #pragma unroll
    for (int j = 0; j < HID_C / 4; ++j) {
        float4 v = hp[j];
        const float* wp = w2s + j * 8;               // rows 4j..4j+3, 2 cols
        a0 += v.x * wp[0]; a1 += v.x * wp[1];
        a0 += v.y * wp[2]; a1 += v.y * wp[3];
        a0 += v.z * wp[4]; a1 += v.z * wp[5];
        a0 += v.w * wp[6]; a1 += v.w * wp[7];
    }
    g2[i * 2 + 0] = a0;
    g2[i * 2 + 1] = a1;
}

// out[i] = b2 + dinv[i]^2 * g2[i]   (self-loop + bias; initializes out)
__global__ void k_selfinit2(const float* __restrict__ g2,
                            const float* __restrict__ dinv,
                            const float* __restrict__ b2,
                            float* __restrict__ out) {
    int i = blockIdx.x * blockDim.x + threadIdx.x;
    if (i >= N_NODES) return;
    float dv = dinv[i];
    float nm = dv * dv;
    out[i * 2 + 0] = b2[0] + nm * g2[i * 2 + 0];
    out[i * 2 + 1] = b2[1] + nm * g2[i * 2 + 1];
}

__global__ void k_scatter2(const int* __restrict__ s32,
                           const int* __restrict__ d32,
                           const float* __restrict__ nrm,
                           const float* __restrict__ g2,
                           float* __restrict__ out) {
    int e = blockIdx.x * blockDim.x + threadIdx.x;
    if (e >= E_EDGES) return;
    int s = s32[e];
    int d = d32[e];
    float nm = nrm[e];
    atomicAdd(out + d * 2 + 0, nm * g2[s * 2 + 0]);
    atomicAdd(out + d * 2 + 1, nm * g2[s * 2 + 1]);
}

// ---------------------------------------------------------------------------

extern "C" void kernel_launch(void* const* d_in, const int* in_sizes, int n_in,
                              void* d_out, int out_size, void* d_ws, size_t ws_size,
                              hipStream_t stream) {
    const float*     x   = (const float*)d_in[0];
    const long long* ei  = (const long long*)d_in[1];
    const float*     ew  = (const float*)d_in[2];
    const float*     W1  = (const float*)d_in[3];
    const float*     b1  = (const float*)d_in[4];
    const float*     W2  = (const float*)d_in[5];
    const float*     b2  = (const float*)d_in[6];
    float*           out = (float*)d_out;

    // Workspace carve-up (256B aligned slices)
    char* w = (char*)d_ws;
    size_t off = 0;
    auto carve = [&](size_t bytes) {
        void* p = w + off;
        off += (bytes + 255) & ~(size_t)255;
        return p;
    };
    float* deg  = (float*)carve(sizeof(float) * N_NODES);          // -> dinv in place
    int*   s32  = (int*)  carve(sizeof(int)   * E_EDGES);
    int*   d32  = (int*)  carve(sizeof(int)   * E_EDGES);
    float* nrm  = (float*)carve(sizeof(float) * E_EDGES);
    float* h    = (float*)carve(sizeof(float) * (size_t)N_NODES * HID_C);
    float* agg  = (float*)carve(sizeof(float) * (size_t)N_NODES * HID_C);
    float* g2   = (float*)carve(sizeof(float) * (size_t)N_NODES * OUT_C);
    (void)ws_size; (void)in_sizes; (void)n_in; (void)out_size;

    const int B = 256;
    const int gN   = (N_NODES + B - 1) / B;
    const int gE   = (E_EDGES + B - 1) / B;
    const int gN32 = (N_NODES * 32 + B - 1) / B;
    const int gE32 = (int)(((long long)E_EDGES * 32 + B - 1) / B);

    // Phase 1: normalization coefficients
    k_init_deg <<<gN, B, 0, stream>>>(deg);
    k_accum_deg<<<gE, B, 0, stream>>>(ei, ew, deg);
    k_dinv     <<<gN, B, 0, stream>>>(deg);
    k_edge_norm<<<gE, B, 0, stream>>>(ei, ew, deg, nrm, s32, d32);

    // Phase 2: h = x @ W1   (WMMA fp32, 2x2 tiles per wave)
    {
        const int mtiles = N_NODES / 16;             // 6250, exact
        dim3 grid((mtiles + 15) / 16, HID_C / 32);   // (391, 4)
        k_gemm1_wmma<<<grid, B, 0, stream>>>(x, W1, h);
    }

    // Phase 3: layer-1 aggregation + bias + ReLU (in agg)
    k_selfinit1<<<gN32, B, 0, stream>>>(h, deg, agg);
    k_scatter1 <<<gE32, B, 0, stream>>>(s32, d32, nrm, h, agg);
    k_bias_relu<<<gN32, B, 0, stream>>>(agg, b1);

    // Phase 4: layer 2
    k_gemm2    <<<gN, B, 0, stream>>>(agg, W2, g2);
    k_selfinit2<<<gN, B, 0, stream>>>(g2, deg, b2, out);
    k_scatter2 <<<gE, B, 0, stream>>>(s32, d32, nrm, g2, out);
}